// GCN_30520037605493
// MI455X (gfx1250) — compile-verified
//
#include <hip/hip_runtime.h>
#include <hip/hip_bf16.h>

#define DFEAT 128

typedef __attribute__((ext_vector_type(2))) float v2f;
typedef __attribute__((ext_vector_type(8))) float v8f;

// ---------------------------------------------------------------------------
// Zero a float buffer (grid-stride).
// ---------------------------------------------------------------------------
__global__ void gcn_zero_f32(float* __restrict__ p, int n) {
  int i = blockIdx.x * blockDim.x + threadIdx.x;
  int stride = gridDim.x * blockDim.x;
  for (; i < n; i += stride) p[i] = 0.0f;
}

// ---------------------------------------------------------------------------
// Degrees: deg_out[row[e]] += 1, deg_in[col[e]] += 1.
// ---------------------------------------------------------------------------
__global__ void gcn_degree(const int* __restrict__ row, const int* __restrict__ col,
                           float* __restrict__ deg_out, float* __restrict__ deg_in,
                           int E) {
  int e = blockIdx.x * blockDim.x + threadIdx.x;
  if (e < E) {
    atomicAdd(&deg_out[row[e]], 1.0f);
    atomicAdd(&deg_in[col[e]], 1.0f);
  }
}

// ---------------------------------------------------------------------------
// x_src = x * rsqrt(max(deg_out, 1)).  One float4 per thread (32 per row).
// ---------------------------------------------------------------------------
__global__ void gcn_scale(const float* __restrict__ x, const float* __restrict__ deg,
                          float* __restrict__ xs, int total4) {
  int t = blockIdx.x * blockDim.x + threadIdx.x;
  if (t < total4) {
    int n = t >> 5;  // D/4 == 32 float4 chunks per row
    float s = rsqrtf(fmaxf(deg[n], 1.0f));
    float4 v = reinterpret_cast<const float4*>(x)[t];
    v.x *= s; v.y *= s; v.z *= s; v.w *= s;
    reinterpret_cast<float4*>(xs)[t] = v;
  }
}

// ---------------------------------------------------------------------------
// Scatter-add: agg[col[e]] += x_src[row[e]].  One wave32 per edge; each lane
// moves a float4 (128 floats total) via global_atomic_add_f32 (L2-resident).
// ---------------------------------------------------------------------------
__global__ void gcn_scatter(const float* __restrict__ xs, const int* __restrict__ row,
                            const int* __restrict__ col, float* __restrict__ agg,
                            int E) {
  int t = blockIdx.x * blockDim.x + threadIdx.x;
  int e = t >> 5;
  int lane = t & 31;
  if (e < E) {
    int r = row[e];
    int c = col[e];
    float4 v = reinterpret_cast<const float4*>(xs + (size_t)r * DFEAT)[lane];
    float* dst = agg + (size_t)c * DFEAT + lane * 4;
    atomicAdd(dst + 0, v.x);
    atomicAdd(dst + 1, v.y);
    atomicAdd(dst + 2, v.z);
    atomicAdd(dst + 3, v.w);
  }
}

// ---------------------------------------------------------------------------
// out = relu?( (agg @ W^T + b) * rsqrt(max(deg_in,1)) [+ resid] )
// FP32 WMMA 16x16x4.  One wave -> one 16x16 output tile.
// Block = 256 thr = 8 waves = one 16-row slab x all 8 column tiles (D=128).
//
// Per ISA 7.12.2 (fp32):
//   A 16x4 : lane m (0-15) VGPR0/1 = K0,K1 ; lanes 16-31 = K2,K3
//   B 4x16 : VGPR r, lanes 0-15 -> K=r ; lanes 16-31 -> K=r+2 ; N = lane&15
//   C 16x16: VGPR r, lanes 0-15 -> M=r ; lanes 16-31 -> M=r+8 ; N = lane&15
// ---------------------------------------------------------------------------
__global__ void __launch_bounds__(256) gcn_gemm_norm(
    const float* __restrict__ A,      // agg [N,128]
    const float* __restrict__ W,      // [128,128] row-major
    const float* __restrict__ bias,   // [128]
    const float* __restrict__ deg_in, // [N]
    const float* __restrict__ resid,  // x (layer 1) or nullptr
    float* __restrict__ out,          // [N,128]
    int do_relu) {
  const int lane = threadIdx.x & 31;
  const int wave = threadIdx.x >> 5;
  const int half = lane >> 4;
  const int l15  = lane & 15;
  const int n0 = blockIdx.x * 16;  // output row tile
  const int i0 = wave * 16;        // output col tile

  const float* ap = A + (size_t)(n0 + l15) * DFEAT + 2 * half;
  const float* bp = W + (size_t)(i0 + l15) * DFEAT + 2 * half;

  v8f acc = {0.f, 0.f, 0.f, 0.f, 0.f, 0.f, 0.f, 0.f};
#pragma unroll
  for (int kk = 0; kk < DFEAT / 4; ++kk) {
    v2f a = *reinterpret_cast<const v2f*>(ap + 4 * kk);
    v2f b = *reinterpret_cast<const v2f*>(bp + 4 * kk);
    // D = A x B + C  (emits v_wmma_f32_16x16x4_f32)
    acc = __builtin_amdgcn_wmma_f32_16x16x4_f32(
        /*neg_a=*/false, a, /*neg_b=*/false, b,
        /*c_mod=*/(short)0, acc, /*reuse_a=*/false, /*reuse_b=*/false);
  }

  const int colI = i0 + l15;
  const float bb = bias[colI];
#pragma unroll
  for (int r = 0; r < 8; ++r) {
    int nrow = n0 + r + 8 * half;
    float val = acc[r] + bb;
    val *= rsqrtf(fmaxf(deg_in[nrow], 1.0f));
    if (resid) val += resid[(size_t)nrow * DFEAT + colI];
    if (do_relu) val = fmaxf(val, 0.0f);
    out[(size_t)nrow * DFEAT + colI] = val;
  }
}

// ---------------------------------------------------------------------------
// Orchestration.  Workspace layout (floats):
//   [0, N)            deg_out
//   [N, 2N)           deg_in
//   [2N, 2N+N*D)      x_src scratch
//   [2N+N*D, 2N+2N*D) agg scratch
// d_out doubles as the layer-1 hidden state (fully overwritten by layer 2).
// ---------------------------------------------------------------------------
extern "C" void kernel_launch(void* const* d_in, const int* in_sizes, int n_in,
                              void* d_out, int out_size, void* d_ws, size_t ws_size,
                              hipStream_t stream) {
  const float* x  = (const float*)d_in[0];
  const int*   ei = (const int*)d_in[1];
  const float* W1 = (const float*)d_in[2];
  const float* b1 = (const float*)d_in[3];
  const float* W2 = (const float*)d_in[4];
  const float* b2 = (const float*)d_in[5];

  const int N = in_sizes[0] / DFEAT;
  const int E = in_sizes[1] / 2;
  const int* row = ei;
  const int* col = ei + E;
  float* out = (float*)d_out;

  float* deg_out = (float*)d_ws;
  float* deg_in  = deg_out + N;
  float* xsrc    = deg_in + N;
  float* agg     = xsrc + (size_t)N * DFEAT;

  const int TB = 256;
  const int ndTotal  = N * DFEAT;
  const int scale4   = N * (DFEAT / 4);
  const long long scatterThreads = (long long)E * 32;

  // degrees (shared by both layers)
  gcn_zero_f32<<<(2 * N + TB - 1) / TB, TB, 0, stream>>>(deg_out, 2 * N);
  gcn_degree<<<(E + TB - 1) / TB, TB, 0, stream>>>(row, col, deg_out, deg_in, E);

  // ---- layer 1: residual + relu ----
  gcn_scale<<<(scale4 + TB - 1) / TB, TB, 0, stream>>>(x, deg_out, xsrc, scale4);
  gcn_zero_f32<<<(ndTotal + TB - 1) / TB, TB, 0, stream>>>(agg, ndTotal);
  gcn_scatter<<<(int)((scatterThreads + TB - 1) / TB), TB, 0, stream>>>(xsrc, row, col, agg, E);
  gcn_gemm_norm<<<N / 16, TB, 0, stream>>>(agg, W1, b1, deg_in, x, out, 1);

  // ---- layer 2: no residual, no relu ----
  gcn_scale<<<(scale4 + TB - 1) / TB, TB, 0, stream>>>(out, deg_out, xsrc, scale4);
  gcn_zero_f32<<<(ndTotal + TB - 1) / TB, TB, 0, stream>>>(agg, ndTotal);
  gcn_scatter<<<(int)((scatterThreads + TB - 1) / TB), TB, 0, stream>>>(xsrc, row, col, agg, E);
  gcn_gemm_norm<<<N / 16, TB, 0, stream>>>(agg, W2, b2, deg_in, nullptr, out, 0);
}